// YOLOH_Expand_66331474920202
// MI455X (gfx1250) — compile-verified
//
#include <hip/hip_runtime.h>
#include <cstdint>

// ---------------- problem constants ----------------
#define HW_M        360000            // H*W*A anchors
#define NCLS        80
#define TOPK        1000
#define LOGIT_THR  (-2.9444389791664403f)   // log(0.05/0.95)  (sigmoid(x)>=0.05)
#define SCALE_CLAMP (4.135166556742356f)    // log(1000/16)
#define CTR_CLAMP   32.0f
#define NMS_THRESH  0.6f

// ---------------- kernel-1 tiling ----------------
#define TILE    128                        // anchors per tile
#define TBYTES  (TILE * NCLS * 4)          // 40960 B contiguous per tile
#define CPITERS ((TBYTES / 16) / 256)      // 10 async b128 issues per thread per tile
#define ROWF    84                         // padded row: 84 floats = 336 B (21 x 16B)
#define ROWB    (ROWF * 4)

// ---------------- selection ----------------
#define NBINS    1024
#define HLO     (-4.0f)
#define HHI     ( 4.0f)
#define CAND_MAX 4096
#define SORT_N   4096

// =====================================================================
// Kernel 1: stream cls_pred through async-LDS double buffer, per-anchor
// max/argmax over 80 logits, write max-logit + class, build histogram.
// =====================================================================
__global__ __launch_bounds__(256) void k_scores(const float* __restrict__ cls_pred,
                                                float* __restrict__ logitOut,
                                                int*   __restrict__ clsOut,
                                                int*   __restrict__ ghist,
                                                int numTiles)
{
    __shared__ __align__(16) unsigned char sbuf[2][TILE * ROWB];   // 2 x 43008 B
    __shared__ float pmax[256];
    __shared__ int   parg[256];
    __shared__ int   shist[NBINS];

    const int tid = threadIdx.x;
    for (int i = tid; i < NBINS; i += 256) shist[i] = 0;

    const long long totalBytes = (long long)HW_M * NCLS * 4;
    const unsigned ldsBase0 = (unsigned)(unsigned long long)(uintptr_t)&sbuf[0][0];
    const unsigned ldsBase1 = (unsigned)(unsigned long long)(uintptr_t)&sbuf[1][0];

    // Issue one tile (40960 contiguous global bytes) into an LDS buffer with
    // 320B->336B row padding.  10 async b128 instructions per wave; counts are
    // uniform across waves because OOB chunks are clamped, never skipped.
    auto issue = [&](int tile, unsigned base) {
        #pragma unroll
        for (int it = 0; it < CPITERS; ++it) {
            int chunk = it * 256 + tid;                      // 16B chunk index
            long long gb = (long long)tile * TBYTES + (long long)chunk * 16;
            long long mx = totalBytes - 16;
            if (gb > mx) gb = mx;                            // clamp tail tile
            const unsigned char* gp = (const unsigned char*)cls_pred + gb;
            int row = chunk / 20;                            // 20 chunks per anchor row
            int col = chunk - row * 20;
            unsigned laddr = base + (unsigned)(row * ROWB + col * 16);
            asm volatile("global_load_async_to_lds_b128 %0, %1, off"
                         :: "v"(laddr), "v"(gp) : "memory");
        }
    };

    int cur = (int)blockIdx.x;
    int sel = 0;
    if (cur < numTiles) issue(cur, ldsBase0);

    while (cur < numTiles) {
        int nxt = cur + (int)gridDim.x;
        if (nxt < numTiles) {
            issue(nxt, sel ? ldsBase0 : ldsBase1);           // prefetch next tile
            asm volatile("s_wait_asynccnt 0xA" ::: "memory"); // wait current tile (<=10 left)
        } else {
            asm volatile("s_wait_asynccnt 0x0" ::: "memory");
        }
        __syncthreads();

        // reduce current tile: 2 threads per anchor (40 classes each)
        const unsigned char* buf = sbuf[sel];
        int a = tid & (TILE - 1);
        int h = tid >> 7;                                     // 0 or 1
        const float* rowp = (const float*)(buf + a * ROWB) + h * 40;
        float m = -3.4e38f; int arg = 0;
        #pragma unroll
        for (int j = 0; j < 40; j += 4) {
            float4 v = *(const float4*)(rowp + j);
            if (v.x > m) { m = v.x; arg = h * 40 + j + 0; }
            if (v.y > m) { m = v.y; arg = h * 40 + j + 1; }
            if (v.z > m) { m = v.z; arg = h * 40 + j + 2; }
            if (v.w > m) { m = v.w; arg = h * 40 + j + 3; }
        }
        pmax[tid] = m; parg[tid] = arg;
        __syncthreads();

        if (tid < TILE) {
            float m0 = pmax[tid];       int a0 = parg[tid];
            float m1 = pmax[tid + 128]; int a1 = parg[tid + 128];
            float fm = (m1 > m0) ? m1 : m0;                  // ties -> lower class idx
            int   fa = (m1 > m0) ? a1 : a0;
            int g = cur * TILE + tid;
            if (g < HW_M) {
                logitOut[g] = fm;
                clsOut[g]   = fa;
                if (fm >= LOGIT_THR) {                       // only valid anchors counted
                    int b = (int)((fm - HLO) * ((float)NBINS / (HHI - HLO)));
                    if (b < 0) b = 0; if (b > NBINS - 1) b = NBINS - 1;
                    atomicAdd(&shist[b], 1);
                }
            }
        }
        __syncthreads();   // everyone done reading sbuf[sel] before it is re-issued
        cur = nxt; sel ^= 1;
    }
    __syncthreads();
    for (int i = tid; i < NBINS; i += 256)
        if (shist[i]) atomicAdd(&ghist[i], shist[i]);
}

// =====================================================================
// Kernel 2: find histogram bin where top-down cumulative count >= TOPK
// =====================================================================
__global__ void k_thresh(const int* __restrict__ ghist, float* __restrict__ thrOut)
{
    if (threadIdx.x == 0 && blockIdx.x == 0) {
        int cum = 0, bin = 0;
        for (int b = NBINS - 1; b >= 0; --b) {
            cum += ghist[b];
            if (cum >= TOPK) { bin = b; break; }
        }
        float edge = HLO + (HHI - HLO) * ((float)bin / (float)NBINS) - 1e-6f;
        if (edge < LOGIT_THR) edge = LOGIT_THR;
        thrOut[0] = edge;
    }
}

// =====================================================================
// Kernel 3: compact candidates >= threshold as sortable u64 keys
//           key = ~monotone(score) << 32 | index  (asc sort == score desc, idx asc)
// =====================================================================
__global__ void k_compact(const float* __restrict__ logit, const float* __restrict__ thr,
                          unsigned long long* __restrict__ cand, int* __restrict__ cnt)
{
    int i = blockIdx.x * blockDim.x + threadIdx.x;
    if (i >= HW_M) return;
    float x = logit[i];
    if (x >= thr[0]) {
        int p = atomicAdd(cnt, 1);
        if (p < CAND_MAX) {
            unsigned fb   = __float_as_uint(x);
            unsigned mono = fb ^ ((unsigned)((int)fb >> 31) | 0x80000000u);
            unsigned key  = ~mono;
            cand[p] = ((unsigned long long)key << 32) | (unsigned)i;
        }
    }
}

// =====================================================================
// Kernel 4: one-block bitonic sort of 4096 keys; gather top-K, apply
// sigmoid only here, decode boxes only for the K selected anchors.
// =====================================================================
__global__ __launch_bounds__(1024) void k_sort_gather(
        const unsigned long long* __restrict__ cand, const int* __restrict__ cnt,
        const float* __restrict__ logit, const int* __restrict__ clsArr,
        const float* __restrict__ reg,   const float* __restrict__ anch,
        float* __restrict__ boxesK, float* __restrict__ scoresK,
        int* __restrict__ clsK, int* __restrict__ validK)
{
    __shared__ unsigned long long keys[SORT_N];   // 32 KB
    int n = *cnt; if (n > CAND_MAX) n = CAND_MAX;
    for (int i = threadIdx.x; i < SORT_N; i += 1024)
        keys[i] = (i < n) ? cand[i] : 0xFFFFFFFFFFFFFFFFull;
    __syncthreads();

    for (int k = 2; k <= SORT_N; k <<= 1) {
        for (int j = k >> 1; j > 0; j >>= 1) {
            for (int i = threadIdx.x; i < SORT_N; i += 1024) {
                int ixj = i ^ j;
                if (ixj > i) {
                    bool up = ((i & k) == 0);
                    unsigned long long a = keys[i], b = keys[ixj];
                    if ((a > b) == up) { keys[i] = b; keys[ixj] = a; }
                }
            }
            __syncthreads();
        }
    }

    int kn = (n < TOPK) ? n : TOPK;
    for (int kk = threadIdx.x; kk < TOPK; kk += 1024) {
        if (kk < kn) {
            int idx = (int)(keys[kk] & 0xFFFFFFFFull);
            float lg = logit[idx];
            float s  = 1.0f / (1.0f + __expf(-lg));
            float rx = reg[4*idx+0], ry = reg[4*idx+1], rw = reg[4*idx+2], rh = reg[4*idx+3];
            float ax = anch[4*idx+0], ay = anch[4*idx+1], aw = anch[4*idx+2], ah = anch[4*idx+3];
            float ox = fminf(fmaxf(rx * aw, -CTR_CLAMP), CTR_CLAMP);
            float oy = fminf(fmaxf(ry * ah, -CTR_CLAMP), CTR_CLAMP);
            float cx = ax + ox, cy = ay + oy;
            float w2 = 0.5f * aw * __expf(fminf(rw, SCALE_CLAMP));
            float h2 = 0.5f * ah * __expf(fminf(rh, SCALE_CLAMP));
            boxesK[4*kk+0] = cx - w2; boxesK[4*kk+1] = cy - h2;
            boxesK[4*kk+2] = cx + w2; boxesK[4*kk+3] = cy + h2;
            scoresK[kk] = s; clsK[kk] = clsArr[idx]; validK[kk] = 1;
        } else {
            boxesK[4*kk+0] = 0.f; boxesK[4*kk+1] = 0.f;
            boxesK[4*kk+2] = 0.f; boxesK[4*kk+3] = 0.f;
            scoresK[kk] = 0.f; clsK[kk] = 0; validK[kk] = 0;
        }
    }
}

// =====================================================================
// Kernel 5: 1000x1000 suppression bit-matrix (same-class & IoU > 0.6)
// one thread per 32-bit word of a row.
// =====================================================================
__global__ void k_supmat(const float* __restrict__ boxesK, const int* __restrict__ clsK,
                         unsigned* __restrict__ mat)
{
    int gid = blockIdx.x * blockDim.x + threadIdx.x;
    if (gid >= TOPK * 32) return;
    int i = gid >> 5, w = gid & 31;
    float x1 = boxesK[4*i+0], y1 = boxesK[4*i+1], x2 = boxesK[4*i+2], y2 = boxesK[4*i+3];
    float ai = (x2 - x1) * (y2 - y1);
    int ci = clsK[i];
    unsigned bits = 0;
    int j0 = w * 32;
    #pragma unroll 4
    for (int jj = 0; jj < 32; ++jj) {
        int j = j0 + jj;
        if (j >= TOPK) break;
        float bx1 = boxesK[4*j+0], by1 = boxesK[4*j+1], bx2 = boxesK[4*j+2], by2 = boxesK[4*j+3];
        float xx1 = fmaxf(x1, bx1), yy1 = fmaxf(y1, by1);
        float xx2 = fminf(x2, bx2), yy2 = fminf(y2, by2);
        float ww = fmaxf(1e-28f, xx2 - xx1);
        float hh = fmaxf(1e-28f, yy2 - yy1);
        float inter = ww * hh;
        float aj = (bx2 - bx1) * (by2 - by1);
        float iou = inter / (ai + aj - inter + 1e-14f);
        if (clsK[j] == ci && iou > NMS_THRESH) bits |= (1u << jj);
    }
    mat[i * 32 + w] = bits;
}

// =====================================================================
// Kernel 6: serial greedy scan by one wave32 (bitmask rows in LDS),
// then write the four flattened outputs.
// =====================================================================
__global__ __launch_bounds__(1024) void k_nms_out(
        const unsigned* __restrict__ mat, const float* __restrict__ boxesK,
        const float* __restrict__ scoresK, const int* __restrict__ clsK,
        const int* __restrict__ validK, float* __restrict__ out)
{
    __shared__ unsigned smat[TOPK * 32];      // 128 KB, L2->LDS staged
    __shared__ unsigned char keepSh[TOPK];
    for (int i = threadIdx.x; i < TOPK * 32; i += 1024) smat[i] = mat[i];
    __syncthreads();

    if (threadIdx.x < 32) {                   // single wave32 does the serial scan
        int lane = threadIdx.x;
        unsigned keepw = 0;                   // lane owns keep-bits for j in [lane*32, lane*32+32)
        for (int i = 0; i < TOPK; ++i) {
            int wi = i >> 5;
            unsigned prior = (lane < wi) ? 0xFFFFFFFFu
                           : (lane == wi ? ((1u << (i & 31)) - 1u) : 0u);
            bool pred = (smat[i * 32 + lane] & keepw & prior) != 0u;
            bool sup  = (__ballot((int)pred) != 0ull);
            bool kp   = (!sup) && (validK[i] != 0);
            if (lane == wi && kp) keepw |= (1u << (i & 31));
            if (lane == 0) keepSh[i] = kp ? 1 : 0;
        }
    }
    __syncthreads();

    // outputs: [0,4000) boxes | [4000,5000) score*keep | [5000,6000) cls i32 | [6000,7000) keep
    for (int k = threadIdx.x; k < TOPK; k += 1024) {
        int kp = keepSh[k];
        out[4*k+0] = boxesK[4*k+0]; out[4*k+1] = boxesK[4*k+1];
        out[4*k+2] = boxesK[4*k+2]; out[4*k+3] = boxesK[4*k+3];
        out[4000 + k] = kp ? scoresK[k] : 0.0f;
        ((int*)out)[5000 + k] = clsK[k];
        out[6000 + k] = kp ? 1.0f : 0.0f;
    }
}

// =====================================================================
extern "C" void kernel_launch(void* const* d_in, const int* in_sizes, int n_in,
                              void* d_out, int out_size, void* d_ws, size_t ws_size,
                              hipStream_t stream)
{
    const float* cls_pred = (const float*)d_in[0];   // (1, M, 80)
    const float* reg_pred = (const float*)d_in[1];   // (1, M, 4)
    const float* anchors  = (const float*)d_in[2];   // (M, 4)

    char*  ws  = (char*)d_ws;
    size_t off = 0;
    auto alloc = [&](size_t bytes) -> void* {
        void* p = ws + off;
        off = (off + bytes + 255) & ~(size_t)255;
        return p;
    };
    int*   hist   = (int*)  alloc(NBINS * 4);
    int*   cnt    = (int*)  alloc(4);
    float* thr    = (float*)alloc(4);
    float* logit  = (float*)alloc((size_t)HW_M * 4);
    int*   clsA   = (int*)  alloc((size_t)HW_M * 4);
    unsigned long long* cand = (unsigned long long*)alloc((size_t)CAND_MAX * 8);
    float* boxesK  = (float*)alloc((size_t)TOPK * 4 * 4);
    float* scoresK = (float*)alloc((size_t)TOPK * 4);
    int*   clsK    = (int*)  alloc((size_t)TOPK * 4);
    int*   validK  = (int*)  alloc((size_t)TOPK * 4);
    unsigned* mat  = (unsigned*)alloc((size_t)TOPK * 32 * 4);

    hipMemsetAsync(hist, 0, NBINS * 4, stream);
    hipMemsetAsync(cnt,  0, 4,         stream);

    int numTiles = (HW_M + TILE - 1) / TILE;         // 2813
    int nblk = 1024; if (nblk > numTiles) nblk = numTiles;
    k_scores <<<nblk, 256, 0, stream>>>(cls_pred, logit, clsA, hist, numTiles);
    k_thresh <<<1, 64, 0, stream>>>(hist, thr);
    k_compact<<<(HW_M + 255) / 256, 256, 0, stream>>>(logit, thr, cand, cnt);
    k_sort_gather<<<1, 1024, 0, stream>>>(cand, cnt, logit, clsA, reg_pred, anchors,
                                          boxesK, scoresK, clsK, validK);
    k_supmat <<<(TOPK * 32 + 255) / 256, 256, 0, stream>>>(boxesK, clsK, mat);
    k_nms_out<<<1, 1024, 0, stream>>>(mat, boxesK, scoresK, clsK, validK, (float*)d_out);
}